// Post_5437428597464
// MI455X (gfx1250) — compile-verified
//
#include <hip/hip_runtime.h>

typedef float v2f __attribute__((ext_vector_type(2)));
typedef float v8f __attribute__((ext_vector_type(8)));

#define T_STEPS 256
#define BF      4096   // B * F_DIM = 8 * 512
#define NS      80     // n_s
#define NTAU    64     // n_taus
#define NG      20     // NS / 4 (K-groups of 4 for f32 WMMA)
#define SEG_LEN 16
#define NSEG    16     // T_STEPS / SEG_LEN

// LDS layout: per K-group g, two half-rows (h = lane>=16) of 64 float2,
// padded so half 0 and half 1 map to disjoint 32-bank sets.
#define G_STRIDE 160   // float2 per group (2*64 data + 32 pad)
#define H_STRIDE 80    // float2 per half

// ---------------------------------------------------------------------------
// Kernel A: per-segment local scan (zero init): end-state E and alpha-sum.
// Uses exp(-s * sum(alpha)) == prod exp(-s*alpha): only a scalar per row
// is needed to describe the segment's total decay.
// ---------------------------------------------------------------------------
__global__ __launch_bounds__(32) void seg_local_scan(
    const float* __restrict__ fs,      // [T, BF]
    const float* __restrict__ alphas,  // [T, BF]
    const float* __restrict__ s,       // [NS]
    float* __restrict__ E,             // [NSEG, BF, NS]
    float* __restrict__ Asum)          // [NSEG, BF]
{
    const int lane = threadIdx.x;
    const int h    = lane >> 4;
    const int col  = lane & 15;
    const int base = blockIdx.x * 16;
    const int seg  = blockIdx.y;
    const int row  = base + col;
    const int koff = h * 2;

    v2f s_reg[NG];
    v2f st[NG];
    #pragma unroll
    for (int g = 0; g < NG; ++g) {
        int k0 = g * 4 + koff;
        s_reg[g].x = s[k0];
        s_reg[g].y = s[k0 + 1];
        st[g].x = 0.0f;
        st[g].y = 0.0f;
    }

    const int t0 = seg * SEG_LEN;
    float asum  = 0.0f;
    float a_cur = alphas[(size_t)t0 * BF + row];
    float f_cur = fs[(size_t)t0 * BF + row];

    for (int tt = 0; tt < SEG_LEN; ++tt) {
        float a_nxt = 0.0f, f_nxt = 0.0f;
        if (tt + 1 < SEG_LEN) {
            a_nxt = alphas[(size_t)(t0 + tt + 1) * BF + row];
            f_nxt = fs[(size_t)(t0 + tt + 1) * BF + row];
        }
        asum += a_cur;
        #pragma unroll
        for (int g = 0; g < NG; ++g) {
            st[g].x = st[g].x * __expf(-s_reg[g].x * a_cur) + f_cur;
            st[g].y = st[g].y * __expf(-s_reg[g].y * a_cur) + f_cur;
        }
        a_cur = a_nxt;
        f_cur = f_nxt;
    }

    float* Eo = E + ((size_t)seg * BF + row) * NS;
    #pragma unroll
    for (int g = 0; g < NG; ++g) {
        int k0 = g * 4 + koff;
        Eo[k0]     = st[g].x;
        Eo[k0 + 1] = st[g].y;
    }
    if (lane < 16) Asum[(size_t)seg * BF + row] = asum;
}

// ---------------------------------------------------------------------------
// Kernel B: sequential combine across the NSEG segments per (row, s):
// emits each segment's true initial state S[seg] and F_last.
// ---------------------------------------------------------------------------
__global__ __launch_bounds__(256) void seg_combine(
    const float* __restrict__ F0,    // [BF, NS]
    const float* __restrict__ s,     // [NS]
    const float* __restrict__ E,     // [NSEG, BF, NS]
    const float* __restrict__ Asum,  // [NSEG, BF]
    float* __restrict__ S,           // [NSEG, BF, NS]
    float* __restrict__ Flast)       // [BF, NS]
{
    int idx = blockIdx.x * 256 + threadIdx.x;
    if (idx >= BF * NS) return;
    int row = idx / NS;
    int k   = idx - row * NS;
    float sv = s[k];
    float st = F0[idx];
    #pragma unroll
    for (int i = 0; i < NSEG; ++i) {
        S[(size_t)i * BF * NS + idx] = st;
        float d = __expf(-sv * Asum[(size_t)i * BF + row]);
        st = st * d + E[(size_t)i * BF * NS + idx];
    }
    Flast[idx] = st;
}

// ---------------------------------------------------------------------------
// Kernel C: WMMA scan over one 16-step segment with the true initial state.
// ---------------------------------------------------------------------------
__global__ __launch_bounds__(32) void post_scan_wmma_seg(
    const float* __restrict__ fs,      // [T, BF]
    const float* __restrict__ alphas,  // [T, BF]
    const float* __restrict__ S,       // [NSEG, BF, NS]
    const float* __restrict__ s,       // [NS]
    const float* __restrict__ post,    // [NS, NTAU]
    float* __restrict__ til)           // [T, BF, NTAU]
{
    __shared__ v2f bshr[NG * G_STRIDE];

    const int lane = threadIdx.x;
    const int h    = lane >> 4;
    const int col  = lane & 15;
    const int base = blockIdx.x * 16;
    const int seg  = blockIdx.y;

    for (int idx = lane; idx < NG * 128; idx += 32) {
        int g  = idx >> 7;
        int r  = idx & 127;
        int hh = r >> 6;
        int n  = r & 63;
        int k0 = g * 4 + hh * 2;
        v2f v;
        v.x = post[k0 * NTAU + n];
        v.y = post[(k0 + 1) * NTAU + n];
        bshr[g * G_STRIDE + hh * H_STRIDE + n] = v;
    }
    __syncthreads();

    const int row  = base + col;
    const int koff = h * 2;

    const float* Si = S + ((size_t)seg * BF + row) * NS;
    v2f s_reg[NG];
    v2f st[NG];
    #pragma unroll
    for (int g = 0; g < NG; ++g) {
        int k0 = g * 4 + koff;
        s_reg[g].x = s[k0];
        s_reg[g].y = s[k0 + 1];
        st[g].x = Si[k0];
        st[g].y = Si[k0 + 1];
    }

    const v2f* bptr = &bshr[h * H_STRIDE + col];

    const int t0 = seg * SEG_LEN;
    float a_cur = alphas[(size_t)t0 * BF + row];
    float f_cur = fs[(size_t)t0 * BF + row];

    for (int tt = 0; tt < SEG_LEN; ++tt) {
        const int t = t0 + tt;
        float a_nxt = 0.0f, f_nxt = 0.0f;
        if (tt + 1 < SEG_LEN) {
            a_nxt = alphas[(size_t)(t + 1) * BF + row];
            f_nxt = fs[(size_t)(t + 1) * BF + row];
        }

        v8f acc0 = {}, acc1 = {}, acc2 = {}, acc3 = {};

        #pragma unroll
        for (int g = 0; g < NG; ++g) {
            v2f a;
            a.x = st[g].x * __expf(-s_reg[g].x * a_cur) + f_cur;
            a.y = st[g].y * __expf(-s_reg[g].y * a_cur) + f_cur;
            st[g] = a;

            const v2f* bg = bptr + g * G_STRIDE;
            acc0 = __builtin_amdgcn_wmma_f32_16x16x4_f32(false, a, false, bg[0],
                                                         (short)0, acc0, false, false);
            acc1 = __builtin_amdgcn_wmma_f32_16x16x4_f32(false, a, false, bg[16],
                                                         (short)0, acc1, false, false);
            acc2 = __builtin_amdgcn_wmma_f32_16x16x4_f32(false, a, false, bg[32],
                                                         (short)0, acc2, false, false);
            acc3 = __builtin_amdgcn_wmma_f32_16x16x4_f32(false, a, false, bg[48],
                                                         (short)0, acc3, false, false);
        }

        size_t obase = ((size_t)t * BF + base + h * 8) * NTAU + col;
        #pragma unroll
        for (int j = 0; j < 8; ++j) {
            float* o = til + obase + (size_t)j * NTAU;
            o[0]  = acc0[j];
            o[16] = acc1[j];
            o[32] = acc2[j];
            o[48] = acc3[j];
        }

        a_cur = a_nxt;
        f_cur = f_nxt;
    }
}

// ---------------------------------------------------------------------------
// Fallback: single-kernel full-T scan (used if workspace is too small).
// ---------------------------------------------------------------------------
__global__ __launch_bounds__(32) void post_scan_wmma_full(
    const float* __restrict__ fs, const float* __restrict__ alphas,
    const float* __restrict__ F0, const float* __restrict__ s,
    const float* __restrict__ post,
    float* __restrict__ til, float* __restrict__ Flast)
{
    __shared__ v2f bshr[NG * G_STRIDE];

    const int lane = threadIdx.x;
    const int h    = lane >> 4;
    const int col  = lane & 15;
    const int base = blockIdx.x * 16;

    for (int idx = lane; idx < NG * 128; idx += 32) {
        int g  = idx >> 7;
        int r  = idx & 127;
        int hh = r >> 6;
        int n  = r & 63;
        int k0 = g * 4 + hh * 2;
        v2f v;
        v.x = post[k0 * NTAU + n];
        v.y = post[(k0 + 1) * NTAU + n];
        bshr[g * G_STRIDE + hh * H_STRIDE + n] = v;
    }
    __syncthreads();

    const int row  = base + col;
    const int koff = h * 2;

    v2f s_reg[NG];
    v2f st[NG];
    #pragma unroll
    for (int g = 0; g < NG; ++g) {
        int k0 = g * 4 + koff;
        s_reg[g].x = s[k0];
        s_reg[g].y = s[k0 + 1];
        st[g].x = F0[row * NS + k0];
        st[g].y = F0[row * NS + k0 + 1];
    }

    const v2f* bptr = &bshr[h * H_STRIDE + col];

    float a_cur = alphas[row];
    float f_cur = fs[row];

    for (int t = 0; t < T_STEPS; ++t) {
        float a_nxt = 0.0f, f_nxt = 0.0f;
        if (t + 1 < T_STEPS) {
            a_nxt = alphas[(size_t)(t + 1) * BF + row];
            f_nxt = fs[(size_t)(t + 1) * BF + row];
        }

        v8f acc0 = {}, acc1 = {}, acc2 = {}, acc3 = {};

        #pragma unroll
        for (int g = 0; g < NG; ++g) {
            v2f a;
            a.x = st[g].x * __expf(-s_reg[g].x * a_cur) + f_cur;
            a.y = st[g].y * __expf(-s_reg[g].y * a_cur) + f_cur;
            st[g] = a;

            const v2f* bg = bptr + g * G_STRIDE;
            acc0 = __builtin_amdgcn_wmma_f32_16x16x4_f32(false, a, false, bg[0],
                                                         (short)0, acc0, false, false);
            acc1 = __builtin_amdgcn_wmma_f32_16x16x4_f32(false, a, false, bg[16],
                                                         (short)0, acc1, false, false);
            acc2 = __builtin_amdgcn_wmma_f32_16x16x4_f32(false, a, false, bg[32],
                                                         (short)0, acc2, false, false);
            acc3 = __builtin_amdgcn_wmma_f32_16x16x4_f32(false, a, false, bg[48],
                                                         (short)0, acc3, false, false);
        }

        size_t obase = ((size_t)t * BF + base + h * 8) * NTAU + col;
        #pragma unroll
        for (int j = 0; j < 8; ++j) {
            float* o = til + obase + (size_t)j * NTAU;
            o[0]  = acc0[j];
            o[16] = acc1[j];
            o[32] = acc2[j];
            o[48] = acc3[j];
        }

        a_cur = a_nxt;
        f_cur = f_nxt;
    }

    #pragma unroll
    for (int g = 0; g < NG; ++g) {
        int k0 = g * 4 + koff;
        Flast[row * NS + k0]     = st[g].x;
        Flast[row * NS + k0 + 1] = st[g].y;
    }
}

extern "C" void kernel_launch(void* const* d_in, const int* in_sizes, int n_in,
                              void* d_out, int out_size, void* d_ws, size_t ws_size,
                              hipStream_t stream) {
    (void)in_sizes; (void)n_in; (void)out_size;
    const float* fs     = (const float*)d_in[0];
    const float* alphas = (const float*)d_in[1];
    const float* F0     = (const float*)d_in[2];
    const float* s      = (const float*)d_in[3];
    const float* post   = (const float*)d_in[4];

    float* til   = (float*)d_out;                       // [T,B,F,NTAU]
    float* Flast = til + (size_t)T_STEPS * BF * NTAU;   // [B,F,NS]

    const size_t nE    = (size_t)NSEG * BF * NS;  // 5.24M floats
    const size_t nAsum = (size_t)NSEG * BF;       // 65K floats
    const size_t need  = (2 * nE + nAsum) * sizeof(float);  // ~42.2 MB

    if (ws_size >= need) {
        float* E    = (float*)d_ws;
        float* Asum = E + nE;
        float* S    = Asum + nAsum;

        dim3 gridA(BF / 16, NSEG);
        seg_local_scan<<<gridA, 32, 0, stream>>>(fs, alphas, s, E, Asum);

        seg_combine<<<(BF * NS + 255) / 256, 256, 0, stream>>>(F0, s, E, Asum, S, Flast);

        dim3 gridC(BF / 16, NSEG);
        post_scan_wmma_seg<<<gridC, 32, 0, stream>>>(fs, alphas, S, s, post, til);
    } else {
        post_scan_wmma_full<<<BF / 16, 32, 0, stream>>>(fs, alphas, F0, s, post, til, Flast);
    }
}